// DotPredictor_78159814853165
// MI455X (gfx1250) — compile-verified
//
#include <hip/hip_runtime.h>
#include <hip/hip_bf16.h>

typedef __attribute__((ext_vector_type(2))) float v2f;
typedef __attribute__((ext_vector_type(8))) float v8f;

#define D_FEAT 64

// One wave32 computes 16 edge scores via 16 chained V_WMMA_F32_16X16X4_F32:
//   A[m][k] = h[src[e0+m]][4c+k]   (16x4 fp32 slab, c = 0..15)
//   B[k][n] = h[dst[e0+n]][4c+k]   (4x16 fp32 slab)
//   D accumulates A*B over c; diag(D)[m] = dot(h[src[e0+m]], h[dst[e0+m]])
__global__ __launch_bounds__(256) void
udotv_wmma_kernel(const float* __restrict__ h,
                  const int*   __restrict__ src,
                  const int*   __restrict__ dst,
                  float*       __restrict__ out,
                  int n_edges)
{
    const int lane        = threadIdx.x & 31;
    const int waveInBlock = threadIdx.x >> 5;
    const int tile        = blockIdx.x * (blockDim.x >> 5) + waveInBlock;
    const int e0          = tile * 16;
    if (e0 >= n_edges) return;          // wave-uniform: EXEC stays all-1s

    // A-matrix 16x4 fp32 VGPR layout (ISA 7.12.2):
    //   lanes 0-15 : VGPR0 = A[lane][0], VGPR1 = A[lane][1]
    //   lanes 16-31: VGPR0 = A[lane-16][2], VGPR1 = A[lane-16][3]
    // B (4x16) uses the mirrored mapping, so A/B gather code is identical.
    const int r  = lane & 15;          // which of the 16 edges this lane serves
    const int kk = (lane >> 4) << 1;   // K sub-pair {kk, kk+1} within each K=4 chunk

    int eL = e0 + r;
    if (eL >= n_edges) eL = n_edges - 1;   // clamp, don't predicate (EXEC must stay full)

    const float* __restrict__ srcRow = h + (long)src[eL] * D_FEAT;
    const float* __restrict__ dstRow = h + (long)dst[eL] * D_FEAT;

    v8f acc = {};
    #pragma unroll
    for (int c = 0; c < 16; ++c) {
        v2f a = *(const v2f*)(srcRow + c * 4 + kk);   // 8B per lane -> row read once
        v2f b = *(const v2f*)(dstRow + c * 4 + kk);
        // 8 args: (neg_a, A, neg_b, B, c_mod, C, reuse_a, reuse_b)
        acc = __builtin_amdgcn_wmma_f32_16x16x4_f32(
                  false, a, false, b, (short)0, acc, false, false);
    }

    // Diagonal extraction (C/D 16x16 fp32 layout):
    //   score[v]   -> VGPR v, lane v        (v = 0..7)
    //   score[v+8] -> VGPR v, lane v + 24
    float val = 0.0f;
    int   m   = -1;
    #pragma unroll
    for (int v = 0; v < 8; ++v) {
        if (lane == v)      { val = acc[v]; m = v;     }
        if (lane == v + 24) { val = acc[v]; m = v + 8; }
    }
    if (m >= 0) {
        int e = e0 + m;
        if (e < n_edges) out[e] = val;
    }
}

extern "C" void kernel_launch(void* const* d_in, const int* in_sizes, int n_in,
                              void* d_out, int out_size, void* d_ws, size_t ws_size,
                              hipStream_t stream) {
    const float* h   = (const float*)d_in[0];
    const int*   src = (const int*)d_in[1];
    const int*   dst = (const int*)d_in[2];
    float*       out = (float*)d_out;

    const int n_edges = in_sizes[1];            // 1,600,000
    const int tiles   = (n_edges + 15) / 16;    // 16 edges per wave
    const int wavesPerBlock = 8;                // 256 threads, wave32
    const int blocks  = (tiles + wavesPerBlock - 1) / wavesPerBlock;

    udotv_wmma_kernel<<<blocks, 256, 0, stream>>>(h, src, dst, out, n_edges);
}